// MyNet_24876450579093
// MI455X (gfx1250) — compile-verified
//
#include <hip/hip_runtime.h>
#include <hip/hip_bf16.h>

typedef __attribute__((ext_vector_type(2))) float v2f;
typedef __attribute__((ext_vector_type(8))) float v8f;

// ---------------------------------------------------------------------------
// Kernel 1: one (sample, branch) per block.
//   conv1 (1->64, K) + relu into padded LDS, then conv2 (64->128, K) as
//   implicit GEMM with V_WMMA_F32_16X16X4_F32, fused relu + avgpool(10).
// ---------------------------------------------------------------------------
template <int K>
__global__ __launch_bounds__(128) void branch_kernel(
    const float* __restrict__ x,       // (512, 200)
    const float* __restrict__ xmask,   // (512)
    const float* __restrict__ w1,      // (64, 1, K)
    const float* __restrict__ b1,      // (64)
    const float* __restrict__ w2,      // (128, 64, K)
    const float* __restrict__ b2,      // (128)
    float* __restrict__ pooled,        // (512, 3840)
    int branch)
{
    constexpr int P = (K - 1) / 2;
    constexpr int Ktot = 64 * K;       // GEMM K dimension (ci*K + dk order)
    constexpr int W = 216;             // padded o1 row width (covers l+dk <= 213)

    __shared__ float xs[208];          // x with halo, data at offset 3
    __shared__ float o1s[64 * W];      // conv1 output, zero padded
    __shared__ float pool[128 * 10];   // per-block pooled accumulator

    const int bn  = blockIdx.x;
    const int tid = threadIdx.x;

    for (int i = tid; i < 208; i += 128) xs[i] = 0.0f;
    for (int i = tid; i < 64 * W; i += 128) o1s[i] = 0.0f;
    for (int i = tid; i < 1280; i += 128) pool[i] = 0.0f;
    __syncthreads();
    for (int i = tid; i < 200; i += 128) xs[3 + i] = x[bn * 200 + i];
    __syncthreads();

    // conv1 + bias + relu -> o1s[ci][P + l]
    for (int idx = tid; idx < 64 * 200; idx += 128) {
        int ci = idx / 200;
        int l  = idx - ci * 200;
        float acc = b1[ci];
#pragma unroll
        for (int dk = 0; dk < K; ++dk)
            acc += w1[ci * K + dk] * xs[3 + l + dk - P];
        o1s[ci * W + P + l] = fmaxf(acc, 0.0f);
    }
    __syncthreads();

    // conv2 as GEMM: M=128 (co, 8 tiles), N=208 (l, 13 tiles), K=Ktot
    const int wave = tid >> 5;
    const int lane = tid & 31;
    const int lm   = lane & 15;
    const int g    = lane >> 4;

    for (int t = wave; t < 8 * 13; t += 4) {
        int mt = t / 13;
        int nt = t - mt * 13;
        v8f c = {};
        // A: w2 row (mt*16+lm), elements kkb + 2*g + {0,1}  (contiguous b64)
        const float* aptr = w2 + (mt * 16 + lm) * Ktot + 2 * g;
        for (int kkb = 0; kkb < Ktot; kkb += 4) {
            v2f a = *(const v2f*)(aptr + kkb);
            int kk0 = kkb + 2 * g;
            int ci0 = kk0 / K, dk0 = kk0 - ci0 * K;
            int kk1 = kk0 + 1;
            int ci1 = kk1 / K, dk1 = kk1 - ci1 * K;
            v2f bfr;
            bfr.x = o1s[ci0 * W + nt * 16 + lm + dk0];
            bfr.y = o1s[ci1 * W + nt * 16 + lm + dk1];
            c = __builtin_amdgcn_wmma_f32_16x16x4_f32(
                    false, a, false, bfr, (short)0, c, false, false);
        }
        // C layout: m = mt*16 + r + 8*g, l = nt*16 + lm
        int l = nt * 16 + lm;
        if (l < 200) {
            int seg = l / 20;
#pragma unroll
            for (int r = 0; r < 8; ++r) {
                int co = mt * 16 + r + 8 * g;
                float v = fmaxf(c[r] + b2[co], 0.0f) * (1.0f / 20.0f);
                atomicAdd(&pool[co * 10 + seg], v);   // ds_add_f32
            }
        }
    }
    __syncthreads();

    float mk = xmask[bn];
    for (int i = tid; i < 1280; i += 128)
        pooled[bn * 3840 + branch * 1280 + i] = pool[i] * mk;
}

// ---------------------------------------------------------------------------
// Kernel 2: xfeat = pooled(512x3840) @ w_xfc^T(32x3840) + b, * valid mask.
// One wave per 16x16 tile, f32 WMMA along K=3840.
// ---------------------------------------------------------------------------
__global__ __launch_bounds__(32) void xfc_kernel(
    const float* __restrict__ pooled,  // (512, 3840)
    const float* __restrict__ wxfc,    // (32, 3840)
    const float* __restrict__ bxfc,    // (32)
    const float* __restrict__ xmask,   // (512)
    float* __restrict__ xfeat)         // (512, 32)
{
    const int mt   = blockIdx.x >> 1;
    const int nt   = blockIdx.x & 1;
    const int lane = threadIdx.x;
    const int lm   = lane & 15;
    const int g    = lane >> 4;

    v8f c = {};
    const float* aptr = pooled + (size_t)(mt * 16 + lm) * 3840 + 2 * g;
    const float* bptr = wxfc   + (size_t)(nt * 16 + lm) * 3840 + 2 * g;
    for (int kkb = 0; kkb < 3840; kkb += 4) {
        v2f a   = *(const v2f*)(aptr + kkb);
        v2f bfr = *(const v2f*)(bptr + kkb);
        c = __builtin_amdgcn_wmma_f32_16x16x4_f32(
                false, a, false, bfr, (short)0, c, false, false);
    }
    int n = nt * 16 + lm;
    float bias = bxfc[n];
#pragma unroll
    for (int r = 0; r < 8; ++r) {
        int m = mt * 16 + r + 8 * g;
        xfeat[m * 32 + n] = (c[r] + bias) * xmask[m];
    }
}

// ---------------------------------------------------------------------------
// Kernel 3: element features + fuse linear (60 -> 64), per (b,n) row.
// ---------------------------------------------------------------------------
__global__ __launch_bounds__(64) void fuse_kernel(
    const float* __restrict__ elem_info,  // (512, 7)
    const float* __restrict__ emask,      // (512)
    const float* __restrict__ w_float,    // (16, 5)
    const float* __restrict__ b_float,    // (16)
    const float* __restrict__ atom_emb,   // (95, 8)
    const float* __restrict__ type_emb,   // (6, 4)
    const float* __restrict__ xfeat,      // (512, 32)
    const float* __restrict__ w_fuse,     // (64, 60)
    const float* __restrict__ b_fuse,     // (64)
    float* __restrict__ fused)            // (512, 64)
{
    __shared__ float feat[60];            // [0:32) xfeat, [32:60) efeat
    const int row = blockIdx.x;
    const int tid = threadIdx.x;
    const float mk = emask[row];
    const bool valid_e = (mk >= 0.5f);

    if (tid < 32) feat[tid] = xfeat[row * 32 + tid];
    if (tid < 28) {
        float masked[7];
#pragma unroll
        for (int c = 0; c < 7; ++c) masked[c] = elem_info[row * 7 + c] * mk;
        int an = (int)masked[5];
        int et = (int)masked[6];
        bool take = valid_e && (an >= 1) && (an <= 94);
        int anc = min(max(an, 0), 94);
        int etc = min(max(et, 0), 5);
        float v;
        if (tid < 16) {
            float acc = b_float[tid];
#pragma unroll
            for (int c = 0; c < 5; ++c) acc += masked[c] * w_float[tid * 5 + c];
            v = fmaxf(acc, 0.0f) * (valid_e ? 1.0f : 0.0f);
        } else if (tid < 24) {
            v = atom_emb[anc * 8 + (tid - 16)] * (take ? 1.0f : 0.0f);
        } else {
            v = type_emb[etc * 4 + (tid - 24)] * (take ? 1.0f : 0.0f);
        }
        feat[32 + tid] = v * mk;
    }
    __syncthreads();

    float acc = b_fuse[tid];
#pragma unroll 4
    for (int c = 0; c < 60; ++c) acc += feat[c] * w_fuse[tid * 60 + c];
    fused[row * 64 + tid] = acc * mk;
}

// ---------------------------------------------------------------------------
// Kernel 4: 4-head attention (N=16, E=64) + masked mean + final MLP, per b.
// ---------------------------------------------------------------------------
__global__ __launch_bounds__(128) void attn_kernel(
    const float* __restrict__ fused,   // (32, 16, 64)
    const float* __restrict__ emask,   // (32, 16)
    const float* __restrict__ in_w,    // (192, 64)
    const float* __restrict__ in_b,    // (192)
    const float* __restrict__ out_w,   // (64, 64)
    const float* __restrict__ out_b,   // (64)
    const float* __restrict__ w_fc1,   // (64, 64)
    const float* __restrict__ b_fc1,   // (64)
    const float* __restrict__ w_fc2,   // (1, 64)
    const float* __restrict__ b_fc2,   // (1)
    float* __restrict__ out)           // (32, 1)
{
    __shared__ float f[16 * 64];
    __shared__ float qkv[16 * 192];
    __shared__ float att[4 * 16 * 16];
    __shared__ float ao[16 * 64];
    __shared__ float aout[16 * 64];
    __shared__ float pooled2[64];
    __shared__ float h1[64];
    __shared__ float msum;

    const int b   = blockIdx.x;
    const int tid = threadIdx.x;

    for (int i = tid; i < 16 * 64; i += 128) f[i] = fused[b * 1024 + i];
    __syncthreads();

    // qkv projection
    for (int i = tid; i < 16 * 192; i += 128) {
        int n = i / 192, j = i - n * 192;
        float acc = in_b[j];
#pragma unroll 8
        for (int c = 0; c < 64; ++c) acc += f[n * 64 + c] * in_w[j * 64 + c];
        qkv[n * 192 + j] = acc;
    }
    __syncthreads();

    // scores (dh=16 -> scale 1/4) + key mask
    for (int i = tid; i < 4 * 16 * 16; i += 128) {
        int h = i >> 8, rem = i & 255;
        int qi = rem >> 4, kj = rem & 15;
        float acc = 0.0f;
#pragma unroll
        for (int d = 0; d < 16; ++d)
            acc += qkv[qi * 192 + h * 16 + d] * qkv[kj * 192 + 64 + h * 16 + d];
        acc *= 0.25f;
        if (emask[b * 16 + kj] < 0.5f) acc = -1e30f;
        att[i] = acc;
    }
    __syncthreads();

    // softmax over the 64 (h, qi) rows
    if (tid < 64) {
        float* rowp = &att[tid * 16];
        float mx = rowp[0];
#pragma unroll
        for (int j = 1; j < 16; ++j) mx = fmaxf(mx, rowp[j]);
        float s = 0.0f;
#pragma unroll
        for (int j = 0; j < 16; ++j) { float e = expf(rowp[j] - mx); rowp[j] = e; s += e; }
        float inv = 1.0f / s;
#pragma unroll
        for (int j = 0; j < 16; ++j) rowp[j] *= inv;
    }
    __syncthreads();

    // attn @ v
    for (int i = tid; i < 16 * 64; i += 128) {
        int qi = i >> 6, e = i & 63;
        int h = e >> 4, d = e & 15;
        float acc = 0.0f;
#pragma unroll
        for (int j = 0; j < 16; ++j)
            acc += att[(h * 16 + qi) * 16 + j] * qkv[j * 192 + 128 + h * 16 + d];
        ao[i] = acc;
    }
    __syncthreads();

    // output projection + mask
    for (int i = tid; i < 16 * 64; i += 128) {
        int qi = i >> 6, e = i & 63;
        float acc = out_b[e];
#pragma unroll 8
        for (int c = 0; c < 64; ++c) acc += ao[qi * 64 + c] * out_w[e * 64 + c];
        aout[i] = acc * emask[b * 16 + qi];
    }
    if (tid == 0) {
        float s = 0.0f;
        for (int j = 0; j < 16; ++j) s += emask[b * 16 + j];
        msum = s + 1e-8f;
    }
    __syncthreads();

    if (tid < 64) {
        float s = 0.0f;
#pragma unroll
        for (int qi = 0; qi < 16; ++qi) s += aout[qi * 64 + tid];
        pooled2[tid] = s / msum;
    }
    __syncthreads();

    if (tid < 64) {
        float acc = b_fc1[tid];
#pragma unroll 8
        for (int c = 0; c < 64; ++c) acc += pooled2[c] * w_fc1[tid * 64 + c];
        h1[tid] = fmaxf(acc, 0.0f);
    }
    __syncthreads();

    if (tid == 0) {
        float acc = b_fc2[0];
        for (int c = 0; c < 64; ++c) acc += h1[c] * w_fc2[c];
        out[b] = acc;
    }
}

// ---------------------------------------------------------------------------
extern "C" void kernel_launch(void* const* d_in, const int* in_sizes, int n_in,
                              void* d_out, int out_size, void* d_ws, size_t ws_size,
                              hipStream_t stream) {
    const float* all_xanes  = (const float*)d_in[0];   // (32,16,200)
    const float* xanes_mask = (const float*)d_in[1];   // (32,16)
    const float* elem_info  = (const float*)d_in[2];   // (32,16,7)
    const float* elem_mask  = (const float*)d_in[3];   // (32,16)
    const float* w_b1c1 = (const float*)d_in[4];
    const float* b_b1c1 = (const float*)d_in[5];
    const float* w_b1c2 = (const float*)d_in[6];
    const float* b_b1c2 = (const float*)d_in[7];
    const float* w_b2c1 = (const float*)d_in[8];
    const float* b_b2c1 = (const float*)d_in[9];
    const float* w_b2c2 = (const float*)d_in[10];
    const float* b_b2c2 = (const float*)d_in[11];
    const float* w_b3c1 = (const float*)d_in[12];
    const float* b_b3c1 = (const float*)d_in[13];
    const float* w_b3c2 = (const float*)d_in[14];
    const float* b_b3c2 = (const float*)d_in[15];
    const float* w_xfc  = (const float*)d_in[16];
    const float* b_xfc  = (const float*)d_in[17];
    const float* w_float = (const float*)d_in[18];
    const float* b_float = (const float*)d_in[19];
    const float* atom_emb = (const float*)d_in[20];
    const float* type_emb = (const float*)d_in[21];
    const float* w_fuse = (const float*)d_in[22];
    const float* b_fuse = (const float*)d_in[23];
    const float* in_w  = (const float*)d_in[24];
    const float* in_b  = (const float*)d_in[25];
    const float* out_w = (const float*)d_in[26];
    const float* out_b = (const float*)d_in[27];
    const float* w_fc1 = (const float*)d_in[28];
    const float* b_fc1 = (const float*)d_in[29];
    const float* w_fc2 = (const float*)d_in[30];
    const float* b_fc2 = (const float*)d_in[31];

    float* ws     = (float*)d_ws;
    float* pooled = ws;                        // 512*3840
    float* xfeat  = pooled + 512 * 3840;       // 512*32
    float* fusedb = xfeat + 512 * 32;          // 512*64

    branch_kernel<3><<<512, 128, 0, stream>>>(all_xanes, xanes_mask,
        w_b1c1, b_b1c1, w_b1c2, b_b1c2, pooled, 0);
    branch_kernel<5><<<512, 128, 0, stream>>>(all_xanes, xanes_mask,
        w_b2c1, b_b2c1, w_b2c2, b_b2c2, pooled, 1);
    branch_kernel<7><<<512, 128, 0, stream>>>(all_xanes, xanes_mask,
        w_b3c1, b_b3c1, w_b3c2, b_b3c2, pooled, 2);

    xfc_kernel<<<64, 32, 0, stream>>>(pooled, w_xfc, b_xfc, xanes_mask, xfeat);

    fuse_kernel<<<512, 64, 0, stream>>>(elem_info, elem_mask, w_float, b_float,
                                        atom_emb, type_emb, xfeat, w_fuse, b_fuse, fusedb);

    attn_kernel<<<32, 128, 0, stream>>>(fusedb, elem_mask, in_w, in_b,
                                        out_w, out_b, w_fc1, b_fc1, w_fc2, b_fc2,
                                        (float*)d_out);
}